// maskedInfoNCE_Loss_79302276153375
// MI455X (gfx1250) — compile-verified
//
#include <hip/hip_runtime.h>
#include <math.h>

#define N      4096
#define BLOCK  256
#define PER_T  (N / BLOCK)   // 16 elements per thread per pass
#define NWAVES (BLOCK / 32)  // 8 waves per block (wave32)

typedef __attribute__((ext_vector_type(2))) float v2f;
typedef __attribute__((ext_vector_type(8))) float v8f;

__device__ __forceinline__ float wave_sum(float v) {
#pragma unroll
  for (int o = 16; o > 0; o >>= 1) v += __shfl_xor(v, o, 32);
  return v;
}

__device__ __forceinline__ float wave_max(float v) {
#pragma unroll
  for (int o = 16; o > 0; o >>= 1) v = fmaxf(v, __shfl_xor(v, o, 32));
  return v;
}

__device__ __forceinline__ float block_sum(float v, float* sred) {
  const int lane = threadIdx.x & 31, wid = threadIdx.x >> 5;
  v = wave_sum(v);
  __syncthreads();                 // protect sred reuse across calls
  if (lane == 0) sred[wid] = v;
  __syncthreads();
  float r = sred[0];
#pragma unroll
  for (int i = 1; i < NWAVES; ++i) r += sred[i];
  return r;
}

__device__ __forceinline__ float block_max(float v, float* sred) {
  const int lane = threadIdx.x & 31, wid = threadIdx.x >> 5;
  v = wave_max(v);
  __syncthreads();
  if (lane == 0) sred[wid] = v;
  __syncthreads();
  float r = sred[0];
#pragma unroll
  for (int i = 1; i < NWAVES; ++i) r = fmaxf(r, sred[i]);
  return r;
}

// One workgroup per row. Row (16KB) + labels (16KB) staged HBM->LDS with
// async-to-LDS b128 (ASYNCcnt), then 3 LDS-resident passes: max, exp+sum,
// masked log1p.  Per-row loss -> row_out[row].
__global__ __launch_bounds__(BLOCK) void row_loss_kernel(
    const float* __restrict__ logits, const int* __restrict__ labels,
    float* __restrict__ row_out) {
  __shared__ float lrow[N];
  __shared__ int   llab[N];
  __shared__ float sred[NWAVES];

  const int row = blockIdx.x;
  const int tid = threadIdx.x;

  // ---- async stage: each thread issues 4+4 b128 loads (16B/lane each) ----
  unsigned ldsRow = (unsigned)(size_t)(&lrow[tid * 4]);
  unsigned ldsLab = (unsigned)(size_t)(&llab[tid * 4]);
  unsigned long long aRow = (unsigned long long)(logits + (size_t)row * N + tid * 4);
  unsigned long long aLab = (unsigned long long)(labels + tid * 4);
  // INST_OFFSET is added to BOTH the LDS address and the global address
  // (CDNA5 ISA), so one base covers all 4 chunks.
  asm volatile(
      "global_load_async_to_lds_b128 %0, %2, off\n\t"
      "global_load_async_to_lds_b128 %0, %2, off offset:4096\n\t"
      "global_load_async_to_lds_b128 %0, %2, off offset:8192\n\t"
      "global_load_async_to_lds_b128 %0, %2, off offset:12288\n\t"
      "global_load_async_to_lds_b128 %1, %3, off\n\t"
      "global_load_async_to_lds_b128 %1, %3, off offset:4096\n\t"
      "global_load_async_to_lds_b128 %1, %3, off offset:8192\n\t"
      "global_load_async_to_lds_b128 %1, %3, off offset:12288\n\t"
      "s_wait_asynccnt 0"
      :: "v"(ldsRow), "v"(ldsLab), "v"(aRow), "v"(aLab)
      : "memory");
  __syncthreads();

  constexpr float SCALE = 7.3890560989306495f;  // e^T, T=2
  constexpr float LOG2E = 1.4426950408889634f;
  constexpr float LN2   = 0.6931471805599453f;

  // ---- pass A: raw row max (scale>0, so max commutes with scaling) ----
  float m = -INFINITY;
#pragma unroll
  for (int k = 0; k < PER_T; ++k) m = fmaxf(m, lrow[tid + k * BLOCK]);
  m = block_max(m, sred);

  // ---- pass B: e = exp(SCALE*(x - m)) via single v_exp_f32; sum ----
  const float kf = SCALE * LOG2E;  // exp(SCALE*(x-m)) = exp2(kf*(x-m))
  float s = 0.0f;
#pragma unroll
  for (int k = 0; k < PER_T; ++k) {
    const int j = tid + k * BLOCK;
    const float e = __builtin_amdgcn_exp2f((lrow[j] - m) * kf);
    lrow[j] = e;  // own element only; re-read by same thread in pass C
    s += e;
  }
  s = block_sum(s, sred);
  const float invS = 1.0f / s;
  const float lnS  = __builtin_amdgcn_logf(s) * LN2;

  // ---- pass C: diag log p_ii + masked log1p(-p_ij) ----
  const int myLab = labels[row];  // scalar (uniform) load
  float acc = 0.0f;
#pragma unroll
  for (int k = 0; k < PER_T; ++k) {
    const int j = tid + k * BLOCK;
    const float e = lrow[j];
    if (j == row) {
      acc += __builtin_amdgcn_logf(e) * LN2 - lnS;  // log p[i,i]
    } else if (llab[j] != myLab) {
      const float p = e * invS;  // in (0,1]
      float l;
      if (p > 0.03f) {
        l = __builtin_amdgcn_logf(1.0f - p) * LN2;
      } else {
        // ln(1-p) = -(p + p^2/2 + p^3/3 + p^4/4) + O(p^5), |err| < 2e-7 rel
        l = -p * (1.0f + p * (0.5f + p * (0.3333333333f + p * 0.25f)));
      }
      acc += l;
    }
  }
  acc = block_sum(acc, sred);
  if (tid == 0) row_out[row] = acc;
}

// Deterministic 4096 -> 1 reduction using V_WMMA_F32_16X16X4_F32 with B=ones:
// D[i][j] = sum_k A[i][k]; summing the full D tile over all lanes recovers the
// exact f32 sum of all 64 A entries regardless of exact A lane layout.
__global__ __launch_bounds__(BLOCK) void reduce_kernel(
    const float* __restrict__ vals, float* __restrict__ out) {
  __shared__ float sred[NWAVES];
  const int tid  = threadIdx.x;
  const int lane = tid & 31, wid = tid >> 5;

  v2f ones; ones.x = 1.0f; ones.y = 1.0f;
  v8f c = {};
  // A is 16x4 f32: lane L<16 holds A[L][0..1] (K=0,1); lane L>=16 holds
  // A[L-16][2..3]; i.e. float index (L%16)*4 + (L>=16 ? 2 : 0), contiguous pair.
  const int colBase = (lane & 15) * 4 + ((lane >> 4) << 1);
#pragma unroll
  for (int t = 0; t < N / (NWAVES * 64); ++t) {  // 8 iterations, 64 vals each
    const float* p = vals + wid * (N / NWAVES) + t * 64 + colBase;
    v2f a; a.x = p[0]; a.y = p[1];
    c = __builtin_amdgcn_wmma_f32_16x16x4_f32(
        /*neg_a=*/false, a, /*neg_b=*/false, ones,
        /*c_mod=*/(short)0, c, /*reuse_a=*/false, /*reuse_b=*/false);
  }
  float v = c[0] + c[1] + c[2] + c[3] + c[4] + c[5] + c[6] + c[7];
  // 32 lanes hold 16 duplicated columns spanning all 16 rows -> /16
  v = wave_sum(v) * (1.0f / 16.0f);
  if (lane == 0) sred[wid] = v;
  __syncthreads();
  if (tid == 0) {
    float r = 0.0f;
#pragma unroll
    for (int i = 0; i < NWAVES; ++i) r += sred[i];
    out[0] = r * (1.0f / (float)N);
  }
}

extern "C" void kernel_launch(void* const* d_in, const int* in_sizes, int n_in,
                              void* d_out, int out_size, void* d_ws, size_t ws_size,
                              hipStream_t stream) {
  const float* logits = (const float*)d_in[0];
  const int*   labels = (const int*)d_in[1];   // JAX x64 disabled -> int32
  float* rowLoss = (float*)d_ws;               // 4096 floats of scratch
  float* outp    = (float*)d_out;

  row_loss_kernel<<<N, BLOCK, 0, stream>>>(logits, labels, rowLoss);
  reduce_kernel<<<1, BLOCK, 0, stream>>>(rowLoss, outp);
}